// GraphSageLayer_6957847019592
// MI455X (gfx1250) — compile-verified
//
#include <hip/hip_runtime.h>

// GraphSAGE layer for MI455X (gfx1250, wave32):
//   c = mean-aggregate(h over incoming edges); out = relu(l2norm([h|c] @ W^T + b))
// K1 zero scratch, K2 edge scatter (L2-resident atomics), K3 fp32 WMMA bundler.

#define D_IN   128
#define D_OUT  128
#define K_TOT  256          // 2 * D_IN
#define ROWSTR 258          // LDS row stride (floats): 258%64==2 -> conflict-free, 8B aligned

typedef __attribute__((ext_vector_type(2))) float v2f;
typedef __attribute__((ext_vector_type(8))) float v8f;

__global__ void sage_zero_ws(float* __restrict__ ws, long n) {
    long i = (long)blockIdx.x * blockDim.x + threadIdx.x;
    long stride = (long)gridDim.x * blockDim.x;
    for (; i < n; i += stride) ws[i] = 0.0f;
}

// One wave32 per edge: lane L handles floats [4L, 4L+4) of the 128-wide row.
__global__ void sage_edge_scatter(const float* __restrict__ h,
                                  const int* __restrict__ src,
                                  const int* __restrict__ dst,
                                  float* __restrict__ summed,
                                  float* __restrict__ deg,
                                  int nedges) {
    long t    = (long)blockIdx.x * blockDim.x + threadIdx.x;
    int  e    = (int)(t >> 5);
    int  lane = (int)(t & 31);
    if (e >= nedges) return;
    int s = src[e];
    int d = dst[e];
    const float4 hv = *(const float4*)(h + (long)s * D_IN + lane * 4);
    float* o = summed + (long)d * D_IN + lane * 4;
    atomicAdd(o + 0, hv.x);
    atomicAdd(o + 1, hv.y);
    atomicAdd(o + 2, hv.z);
    atomicAdd(o + 3, hv.w);
    if (lane == 0) atomicAdd(deg + d, 1.0f);
}

// Block: 256 threads = 8 wave32s. Block owns 16 output rows; wave w owns the
// 16-column tile [16w, 16w+16). A = concat(h, summed/deg) staged in LDS once.
// fp32 WMMA: D(16x16) += A(16x4) x B(4x16), 64 steps over K=256.
__global__ __launch_bounds__(256)
void sage_bundler_wmma(const float* __restrict__ h,
                       const float* __restrict__ summed,
                       const float* __restrict__ deg,
                       const float* __restrict__ W,     // [128, 256] row-major
                       const float* __restrict__ bias,  // [128]
                       float* __restrict__ out,         // [N, 128]
                       int nnodes) {
    __shared__ float sA[16 * ROWSTR];   // 16 rows x 256 (padded) concat tile
    __shared__ float s_invdeg[16];
    __shared__ float s_sumsq[16];

    const int tid   = threadIdx.x;
    const int wave  = tid >> 5;
    const int lane  = tid & 31;
    const int mBase = blockIdx.x * 16;

    if (tid < 16) {
        int row = mBase + tid;
        float dg = (row < nnodes) ? deg[row] : 1.0f;
        s_invdeg[tid] = 1.0f / fmaxf(dg, 1.0f);
        s_sumsq[tid]  = 0.0f;
    }
    __syncthreads();

    // Cooperative LDS fill of the 16x256 concat tile.
    for (int idx = tid; idx < 16 * K_TOT; idx += 256) {
        int  r   = idx >> 8;          // 0..15
        int  k   = idx & 255;         // 0..255
        long row = mBase + r;
        if (row >= nnodes) row = nnodes - 1;   // clamp (grid may overshoot)
        float v;
        if (k < D_IN) v = h[row * D_IN + k];
        else          v = summed[row * D_IN + (k - D_IN)] * s_invdeg[r];
        sA[r * ROWSTR + k] = v;
    }
    __syncthreads();

    // Fragment addressing per ISA 16x16x4 f32 layout:
    //   A: lane holds row M = lane&15; VGPR j holds K = 2*(lane>>4) + j
    //   B: lane holds col N = lane&15; VGPR j holds K = 2*(lane>>4) + j
    const int arow = lane & 15;
    const int col  = wave * 16 + (lane & 15);
    const int kb   = (lane >> 4) * 2;
    const float* wrow = W + (long)col * K_TOT;

    v8f acc = {};
    #pragma unroll 4
    for (int k0 = 0; k0 < K_TOT; k0 += 4) {
        v2f a = *(const v2f*)&sA[arow * ROWSTR + k0 + kb];   // ds_load_b64
        v2f b = *(const v2f*)&wrow[k0 + kb];                 // global_load_b64 (L2-hot)
        acc = __builtin_amdgcn_wmma_f32_16x16x4_f32(
                  /*neg_a=*/false, a, /*neg_b=*/false, b,
                  /*c_mod=*/(short)0, acc, /*reuse_a=*/false, /*reuse_b=*/false);
    }

    // Bias (column-broadcast: fragment column N = lane&15 for all 8 regs).
    float bc = bias[col];
    #pragma unroll
    for (int i = 0; i < 8; ++i) acc[i] += bc;

    // Row sum-of-squares: reduce across the 16-lane N group, then LDS atomic
    // across the 8 waves (each owns a disjoint 16-column slice of the row).
    #pragma unroll
    for (int i = 0; i < 8; ++i) {
        float v = acc[i] * acc[i];
        v += __shfl_xor(v, 1, 32);
        v += __shfl_xor(v, 2, 32);
        v += __shfl_xor(v, 4, 32);
        v += __shfl_xor(v, 8, 32);
        if ((lane & 15) == 0)
            atomicAdd(&s_sumsq[i + ((lane >> 4) << 3)], v);   // ds_add_f32
    }
    __syncthreads();

    // Normalize + ReLU + store. Fragment row M = i + 8*(lane>=16).
    #pragma unroll
    for (int i = 0; i < 8; ++i) {
        int  r     = i + ((lane >> 4) << 3);
        long row   = mBase + r;
        if (row >= nnodes) continue;
        float nrm   = sqrtf(s_sumsq[r]);
        float scale = 1.0f / fmaxf(nrm, 1e-12f);
        float v     = fmaxf(acc[i] * scale, 0.0f);
        out[row * D_OUT + col] = v;
    }
}

extern "C" void kernel_launch(void* const* d_in, const int* in_sizes, int n_in,
                              void* d_out, int out_size, void* d_ws, size_t ws_size,
                              hipStream_t stream) {
    const float* h    = (const float*)d_in[0];
    const float* W    = (const float*)d_in[1];
    const float* bias = (const float*)d_in[2];
    const int*   esrc = (const int*)d_in[3];
    const int*   edst = (const int*)d_in[4];
    float*       out  = (float*)d_out;

    const int nnodes = in_sizes[0] / D_IN;
    const int nedges = in_sizes[3];

    float* summed = (float*)d_ws;                       // [N, 128]
    float* deg    = summed + (long)nnodes * D_IN;       // [N]

    // K1: zero accumulators (N*129 floats).
    long zn = (long)nnodes * (D_IN + 1);
    sage_zero_ws<<<2048, 256, 0, stream>>>(summed, zn);

    // K2: one wave32 per edge.
    long ethreads = (long)nedges * 32;
    sage_edge_scatter<<<(int)((ethreads + 255) / 256), 256, 0, stream>>>(
        h, esrc, edst, summed, deg, nedges);

    // K3: bundler GEMM + L2-normalize + ReLU. 16 rows per block.
    int nblocks = (nnodes + 15) / 16;
    sage_bundler_wmma<<<nblocks, 256, 0, stream>>>(
        h, summed, deg, W, bias, out, nnodes);
}